// LinearNoSum_9053791060273
// MI455X (gfx1250) — compile-verified
//
#include <hip/hip_runtime.h>

// LinearNoSum: out[i, b*O + o] = input[b,i] * exp(weight[i,o]) + bias[i,o]
// I=512, O=128, B=2048.  Pure store-bandwidth-bound (536 MB out @ 23.3 TB/s ~ 23us).
// Implemented as rank-1 outer product per row i via V_WMMA_F32_16X16X4_F32:
//   A (16x4): column K=0 = input[b0+M, i], rest 0
//   B (4x16): row    K=0 = exp(weight[i, o0+N]), rest 0
//   C (16x16): bias[i, o0+N] broadcast down M
//   D = A*B + C  == the exact 16x16 output tile, all-f32, bias add free.

typedef float v2f __attribute__((ext_vector_type(2)));
typedef float v8f __attribute__((ext_vector_type(8)));

#define IN_FEATS  512
#define OUT_FEATS 128
#define BATCH     2048

__global__ __launch_bounds__(256) void linear_nosum_wmma_kernel(
    const float* __restrict__ input,   // [BATCH, IN_FEATS]
    const float* __restrict__ weight,  // [IN_FEATS, OUT_FEATS]
    const float* __restrict__ bias,    // [IN_FEATS, OUT_FEATS]
    float* __restrict__ out)           // [IN_FEATS, BATCH*OUT_FEATS]
{
    const int tid  = threadIdx.x;
    const int wave = tid >> 5;        // 0..7
    const int lane = tid & 31;
    const int half = lane >> 4;       // 0: lanes 0-15, 1: lanes 16-31
    const int n    = lane & 15;       // N (and M offset within half) index

    const int i     = (int)(blockIdx.x >> 4);   // feature row 0..511
    const int chunk = (int)(blockIdx.x & 15);   // batch chunk of 128
    const int b0    = chunk * 128 + wave * 16;  // this wave's 16-row b-tile

    // ---- A matrix (16x4 f32, 2 VGPRs/lane). Only K=0 column is nonzero. ----
    // ISA layout: lanes 0-15 hold K=0 (vgpr0) / K=1 (vgpr1); lanes 16-31 hold K=2/K=3.
    const float ain = input[(size_t)(b0 + n) * IN_FEATS + i]; // A[M=n, K=0]
    v2f a;
    a.x = (half == 0) ? ain : 0.0f;
    a.y = 0.0f;

    const float* wrow = weight + (size_t)i * OUT_FEATS;
    const float* brow = bias   + (size_t)i * OUT_FEATS;
    float*       orow = out    + (size_t)i * ((size_t)BATCH * OUT_FEATS);

#pragma unroll
    for (int ot = 0; ot < 8; ++ot) {           // 8 o-tiles of 16 cover O=128
        const int o0 = ot * 16;

        // ---- B matrix (4x16 f32). Only K=0 row nonzero: exp(weight[i, o0+N]).
        // K=0 row lives in vgpr0, lanes 0-15 under the documented row striping.
        const float ew = expf(wrow[o0 + n]);
        v2f bm;
        bm.x = (half == 0) ? ew : 0.0f;
        bm.y = 0.0f;

        // ---- C matrix: bias[i, o0+N], identical for every row M -> replicate
        // across all 8 C/D VGPRs (VGPR v holds rows M=v and M=v+8).
        const float cv = brow[o0 + n];
        v8f c = {cv, cv, cv, cv, cv, cv, cv, cv};

        // D = A x B + C   (EXEC is all-1s: no divergent control flow anywhere)
        v8f d = __builtin_amdgcn_wmma_f32_16x16x4_f32(
            /*neg_a=*/false, a, /*neg_b=*/false, bm,
            /*c_mod=*/(short)0, c, /*reuse_a=*/false, /*reuse_b=*/false);

        // ---- Store D: VGPR v -> row M = v + 8*half, col N = n.
        // Each store writes two contiguous 64B row segments. Non-temporal:
        // 512MB output stream cannot (and should not) live in L2.
        const size_t base = (size_t)(b0 + half * 8) * OUT_FEATS + (size_t)(o0 + n);
#pragma unroll
        for (int v = 0; v < 8; ++v) {
            __builtin_nontemporal_store(d[v], orow + base + (size_t)v * OUT_FEATS);
        }
    }
}

extern "C" void kernel_launch(void* const* d_in, const int* in_sizes, int n_in,
                              void* d_out, int out_size, void* d_ws, size_t ws_size,
                              hipStream_t stream) {
    const float* input  = (const float*)d_in[0];  // [2048, 512]
    const float* weight = (const float*)d_in[1];  // [512, 128]
    const float* bias   = (const float*)d_in[2];  // [512, 128]
    float*       out    = (float*)d_out;          // [512, 2048*128]

    (void)in_sizes; (void)n_in; (void)out_size; (void)d_ws; (void)ws_size;

    // 512 feature rows x 16 batch-chunks (128 b each) = 8192 blocks,
    // 256 threads = 8 waves; each wave: one 16-row b-tile x all 8 o-tiles.
    dim3 grid(IN_FEATS * (BATCH / 128));
    dim3 block(256);
    linear_nosum_wmma_kernel<<<grid, block, 0, stream>>>(input, weight, bias, out);
}